// TimeSeriesWineSNN_31877247271596
// MI455X (gfx1250) — compile-verified
//
#include <hip/hip_runtime.h>

typedef float v2f __attribute__((ext_vector_type(2)));
typedef float v8f __attribute__((ext_vector_type(8)));

namespace {
constexpr int   T_STEPS = 1000;
constexpr int   BATCH   = 8192;
constexpr int   NI  = 6;
constexpr int   NH1 = 28;
constexpr int   NH2 = 8;
constexpr int   NO  = 3;
constexpr float BETA   = 0.9f;
constexpr float THRESH = 1.0f;
constexpr int   WAVES_PER_BLOCK = 8;
constexpr int   TILE_B = 16;        // batch rows per wave (WMMA M dimension)
constexpr int   SM2_STRIDE = 9;     // pad to dodge LDS bank conflicts
}

static __device__ __forceinline__ v8f wmma_f32(v2f a, v2f b, v8f c) {
    // (neg_a, A, neg_b, B, c_mod, C, reuse_a, reuse_b)
    return __builtin_amdgcn_wmma_f32_16x16x4_f32(false, a, false, b, (short)0, c, false, false);
}

// Transpose staging buffers are wave-private and LDS ops complete in order
// within a wave (ISA ch.5: DScnt counts LDS ops, which return in order), so a full s_barrier is
// unnecessary. This is a compiler code-motion fence + hardware DScnt drain.
static __device__ __forceinline__ void lds_wave_fence() {
    asm volatile("s_wait_dscnt 0x0" ::: "memory");
}

__global__ __launch_bounds__(WAVES_PER_BLOCK * 32)
void snn_fused_kernel(const float* __restrict__ x,
                      const float* __restrict__ W1, const float* __restrict__ b1,
                      const float* __restrict__ W2, const float* __restrict__ b2,
                      const float* __restrict__ W3, const float* __restrict__ b3,
                      float* __restrict__ mem3_out, float* __restrict__ spk3_out,
                      float* __restrict__ spk1_out, float* __restrict__ spk2_out)
{
    const int lane = threadIdx.x & 31;
    const int wib  = threadIdx.x >> 5;
    const int wave = blockIdx.x * WAVES_PER_BLOCK + wib;
    const int b0   = wave * TILE_B;     // batch tile base
    const int m    = lane & 15;         // A: M row / B,C,D: N column
    const int half = lane >> 4;         // lane half selects K (+2) or M (+8)
    const int bloc_base = 8 * half;     // C/D element v -> batch row bloc_base+v

    __shared__ float lds1[WAVES_PER_BLOCK][TILE_B * NH1];
    __shared__ float lds2[WAVES_PER_BLOCK][TILE_B * SM2_STRIDE];
    float* sm1 = lds1[wib];
    float* sm2 = lds2[wib];

    // ---- preload weights once, in WMMA f32 16x16x4 B-layout ----
    // B (4x16): lane = N, vgpr v -> K = v + 2*half
    v2f Wb1[2][2], Wb2[7], Wb3[2];
    #pragma unroll
    for (int nt = 0; nt < 2; ++nt)
      #pragma unroll
      for (int ks = 0; ks < 2; ++ks)
        #pragma unroll
        for (int v = 0; v < 2; ++v) {
            int n = nt * 16 + m, k = ks * 4 + v + 2 * half;
            Wb1[nt][ks][v] = (n < NH1 && k < NI) ? W1[n * NI + k] : 0.0f;
        }
    #pragma unroll
    for (int ks = 0; ks < 7; ++ks)
      #pragma unroll
      for (int v = 0; v < 2; ++v) {
            int k = ks * 4 + v + 2 * half;              // k < 28 always
            Wb2[ks][v] = (m < NH2) ? W2[m * NH1 + k] : 0.0f;
      }
    #pragma unroll
    for (int ks = 0; ks < 2; ++ks)
      #pragma unroll
      for (int v = 0; v < 2; ++v) {
            int k = ks * 4 + v + 2 * half;              // k < 8 always
            Wb3[ks][v] = (m < NO) ? W3[m * NH2 + k] : 0.0f;
      }
    float bias1[2];
    bias1[0] = b1[m];
    bias1[1] = (16 + m < NH1) ? b1[16 + m] : 0.0f;
    const float bias2 = (m < NH2) ? b2[m] : 0.0f;
    const float bias3 = (m < NO)  ? b3[m] : 0.0f;

    // membrane state in WMMA C/D layout: lane = neuron (N), elem v -> batch row
    v8f m1s[2], m2s, m3s;
    #pragma unroll
    for (int v = 0; v < 8; ++v) { m1s[0][v] = 0.f; m1s[1][v] = 0.f; m2s[v] = 0.f; m3s[v] = 0.f; }

    // streaming pointers (advance by constant stride per timestep)
    const float* xptr = x + (size_t)(b0 + m) * NI;
    float* p1  = spk1_out + (size_t)(b0 + bloc_base) * NH1 + m;
    float* p2  = spk2_out + (size_t)(b0 + bloc_base) * NH2 + m;
    float* p3m = mem3_out + (size_t)(b0 + bloc_base) * NO  + m;
    float* p3s = spk3_out + (size_t)(b0 + bloc_base) * NO  + m;

    for (int t = 0; t < T_STEPS; ++t) {
        // ---- A = x tile (16 x 6, K padded to 8), unconditional 8B loads ----
        // xa0: k = 2*half + v (always valid; element index even -> 8B aligned)
        v2f xa0 = *(const v2f*)(xptr + 2 * half);
        // xa1: half0 -> k = 4,5 ; half1 -> dummy in-bounds read of k = 0,1, zeroed
        v2f xraw = *(const v2f*)(xptr + 4 - 4 * half);
        v2f xa1;
        xa1[0] = half ? 0.0f : xraw[0];
        xa1[1] = half ? 0.0f : xraw[1];

        // ---- layer 1, N-tile 0 (neurons 0..15, all lanes valid) ----
        {
            v8f c;
            #pragma unroll
            for (int v = 0; v < 8; ++v) c[v] = bias1[0];
            c = wmma_f32(xa0, Wb1[0][0], c);
            c = wmma_f32(xa1, Wb1[0][1], c);
            v8f mm = m1s[0];
            #pragma unroll
            for (int v = 0; v < 8; ++v) {
                float mold  = mm[v];
                float reset = (mold > THRESH) ? THRESH : 0.0f;
                float mnew  = BETA * mold + c[v] - reset;
                float spk   = (mnew > THRESH) ? 1.0f : 0.0f;
                mm[v] = mnew;
                p1[v * NH1] = spk;
                sm1[(bloc_base + v) * NH1 + m] = spk;
            }
            m1s[0] = mm;
        }
        // ---- layer 1, N-tile 1 (neurons 16..27, valid for m < 12) ----
        {
            v8f c;
            #pragma unroll
            for (int v = 0; v < 8; ++v) c[v] = bias1[1];
            c = wmma_f32(xa0, Wb1[1][0], c);
            c = wmma_f32(xa1, Wb1[1][1], c);
            v8f mm = m1s[1];
            #pragma unroll
            for (int v = 0; v < 8; ++v) {
                float mold  = mm[v];
                float reset = (mold > THRESH) ? THRESH : 0.0f;
                float mnew  = BETA * mold + c[v] - reset;
                float spk   = (mnew > THRESH) ? 1.0f : 0.0f;
                mm[v] = mnew;
                if (m < NH1 - 16) {
                    p1[v * NH1 + 16] = spk;
                    sm1[(bloc_base + v) * NH1 + 16 + m] = spk;
                }
            }
            m1s[1] = mm;
        }
        lds_wave_fence();

        // ---- layer 2: cur2 = spk1(16x28) @ W2^T ----
        v8f c2;
        #pragma unroll
        for (int v = 0; v < 8; ++v) c2[v] = bias2;
        #pragma unroll
        for (int ks = 0; ks < 7; ++ks) {
            v2f a2;
            a2[0] = sm1[m * NH1 + ks * 4 + 0 + 2 * half];
            a2[1] = sm1[m * NH1 + ks * 4 + 1 + 2 * half];
            c2 = wmma_f32(a2, Wb2[ks], c2);
        }
        {
            v8f mm = m2s;
            #pragma unroll
            for (int v = 0; v < 8; ++v) {
                float mold  = mm[v];
                float reset = (mold > THRESH) ? THRESH : 0.0f;
                float mnew  = BETA * mold + c2[v] - reset;
                float spk   = (mnew > THRESH) ? 1.0f : 0.0f;
                mm[v] = mnew;
                if (m < NH2) {
                    p2[v * NH2] = spk;
                    sm2[(bloc_base + v) * SM2_STRIDE + m] = spk;
                }
            }
            m2s = mm;
        }
        lds_wave_fence();

        // ---- layer 3: cur3 = spk2(16x8) @ W3^T ----
        v8f c3;
        #pragma unroll
        for (int v = 0; v < 8; ++v) c3[v] = bias3;
        #pragma unroll
        for (int ks = 0; ks < 2; ++ks) {
            v2f a3;
            a3[0] = sm2[m * SM2_STRIDE + ks * 4 + 0 + 2 * half];
            a3[1] = sm2[m * SM2_STRIDE + ks * 4 + 1 + 2 * half];
            c3 = wmma_f32(a3, Wb3[ks], c3);
        }
        {
            v8f mm = m3s;
            #pragma unroll
            for (int v = 0; v < 8; ++v) {
                float mold  = mm[v];
                float reset = (mold > THRESH) ? THRESH : 0.0f;
                float mnew  = BETA * mold + c3[v] - reset;
                float spk   = (mnew > THRESH) ? 1.0f : 0.0f;
                mm[v] = mnew;
                if (m < NO) {
                    p3m[v * NO] = mnew;     // reference records post-update membrane
                    p3s[v * NO] = spk;
                }
            }
            m3s = mm;
        }

        xptr += (size_t)BATCH * NI;
        p1   += (size_t)BATCH * NH1;
        p2   += (size_t)BATCH * NH2;
        p3m  += (size_t)BATCH * NO;
        p3s  += (size_t)BATCH * NO;
    }
}

extern "C" void kernel_launch(void* const* d_in, const int* in_sizes, int n_in,
                              void* d_out, int out_size, void* d_ws, size_t ws_size,
                              hipStream_t stream) {
    (void)in_sizes; (void)n_in; (void)d_ws; (void)ws_size; (void)out_size;
    const float* x  = (const float*)d_in[0];
    const float* W1 = (const float*)d_in[1];
    const float* b1 = (const float*)d_in[2];
    const float* W2 = (const float*)d_in[3];
    const float* b2 = (const float*)d_in[4];
    const float* W3 = (const float*)d_in[5];
    const float* b3 = (const float*)d_in[6];

    float* out = (float*)d_out;
    const size_t TB = (size_t)T_STEPS * (size_t)BATCH;
    float* mem3 = out;                        // (T,B,3)
    float* spk3 = out + TB * 3;               // (T,B,3)
    float* spk1 = out + TB * 6;               // (T,B,28)
    float* spk2 = out + TB * 6 + TB * NH1;    // (T,B,8)

    dim3 block(WAVES_PER_BLOCK * 32);                    // 256 threads = 8 waves
    dim3 grid(BATCH / (TILE_B * WAVES_PER_BLOCK));       // 64 blocks, 512 waves
    hipLaunchKernelGGL(snn_fused_kernel, grid, block, 0, stream,
                       x, W1, b1, W2, b2, W3, b3, mem3, spk3, spk1, spk2);
}